// Net_76510547411421
// MI455X (gfx1250) — compile-verified
//
#include <hip/hip_runtime.h>
#include <math.h>

#define N_NODES 50000
#define N_EDGES 640000
#define HDIM    128
#define NGRAPH  256
#define NCLASS  10
#define MAXSEG  1024   // per-graph node cap for LDS fast path (actual max ~260)

typedef float v2f __attribute__((ext_vector_type(2)));
typedef float v8f __attribute__((ext_vector_type(8)));

// ---------------- segment bounds: starts/totals per graph (batch is sorted) --
__global__ void seg_bounds_kernel(const int* __restrict__ batch,
                                  int* __restrict__ starts, int* __restrict__ totals) {
  int g = threadIdx.x;                       // 256 threads
  int lo = 0, hi = N_NODES;
  while (lo < hi) { int mid = (lo + hi) >> 1; if (batch[mid] < g) lo = mid + 1; else hi = mid; }
  int s0 = lo;
  lo = 0; hi = N_NODES;
  while (lo < hi) { int mid = (lo + hi) >> 1; if (batch[mid] < g + 1) lo = mid + 1; else hi = mid; }
  starts[g] = s0;
  totals[g] = lo - s0;
}

__global__ void init_mask_kernel(float* __restrict__ maskf) {
  int i = blockIdx.x * blockDim.x + threadIdx.x;
  if (i < N_NODES) maskf[i] = 1.0f;
}

// ================= CSR-by-destination build (topology constant, built once) ==
__global__ void degree_kernel(const int* __restrict__ ei, int* __restrict__ deg) {
  int e = blockIdx.x * blockDim.x + threadIdx.x;
  if (e < N_EDGES) atomicAdd(&deg[ei[N_EDGES + e]], 1);
}

// single-block chunked exclusive scan over N_NODES degrees -> rowptr[N_NODES+1]
__global__ void scan_kernel(const int* __restrict__ deg, int* __restrict__ rowptr) {
  __shared__ int buf[1024];
  __shared__ int carry;
  if (threadIdx.x == 0) carry = 0;
  __syncthreads();
  for (int base = 0; base < N_NODES; base += 1024) {
    int i = base + threadIdx.x;
    int v = (i < N_NODES) ? deg[i] : 0;
    buf[threadIdx.x] = v;
    __syncthreads();
    for (int off = 1; off < 1024; off <<= 1) {
      int add = (threadIdx.x >= off) ? buf[threadIdx.x - off] : 0;
      __syncthreads();
      buf[threadIdx.x] += add;
      __syncthreads();
    }
    if (i < N_NODES) rowptr[i] = carry + buf[threadIdx.x] - v;  // exclusive
    __syncthreads();
    if (threadIdx.x == 0) carry += buf[1023];
    __syncthreads();
  }
  if (threadIdx.x == 0) rowptr[N_NODES] = carry;               // == N_EDGES
}

__global__ void copy_pos_kernel(const int* __restrict__ rowptr, int* __restrict__ pos) {
  int i = blockIdx.x * blockDim.x + threadIdx.x;
  if (i < N_NODES) pos[i] = rowptr[i];
}

__global__ void fill_kernel(const int* __restrict__ ei, int* __restrict__ pos,
                            int* __restrict__ csr_src) {
  int e = blockIdx.x * blockDim.x + threadIdx.x;
  if (e < N_EDGES) {
    int d = ei[N_EDGES + e];
    int idx = atomicAdd(&pos[d], 1);
    csr_src[idx] = ei[e];
  }
}

// insertion-sort each node's bucket by src id -> deterministic summation order
__global__ void sortbucket_kernel(const int* __restrict__ rowptr, int* __restrict__ csr_src) {
  int n = blockIdx.x * blockDim.x + threadIdx.x;
  if (n >= N_NODES) return;
  int s = rowptr[n], e = rowptr[n + 1];
  for (int i = s + 1; i < e; ++i) {
    int v = csr_src[i];
    int j = i - 1;
    while (j >= s && csr_src[j] > v) { csr_src[j + 1] = csr_src[j]; --j; }
    csr_src[j + 1] = v;
  }
}

// ---------------- aggregation as coalesced gather: agg[n] = sum h[src] -------
__global__ void gather_kernel(const int* __restrict__ rowptr, const int* __restrict__ csr_src,
                              const float* __restrict__ h, float* __restrict__ agg) {
  int n = blockIdx.x, f = threadIdx.x;       // 128 threads = HDIM
  int s = rowptr[n], e = rowptr[n + 1];
  float acc = 0.f;
  for (int j = s; j < e; ++j)
    acc += h[(size_t)csr_src[j] * HDIM + f];
  agg[(size_t)n * HDIM + f] = acc;
}

// ---------------- fused GraphConv: out = relu(h@Wr + agg@Wn + b) * mask ------
// One wave computes one 16x16 tile via V_WMMA_F32_16X16X4_F32, K=128 in steps of 4.
__global__ void conv_wmma_kernel(const float* __restrict__ h, const float* __restrict__ agg,
                                 const float* __restrict__ Wr, const float* __restrict__ Wn,
                                 const float* __restrict__ bias, const float* __restrict__ maskf,
                                 float* __restrict__ out) {
  const int wave = threadIdx.x >> 5;         // 8 waves -> 8 column tiles
  const int lane = threadIdx.x & 31;
  const int row0 = blockIdx.x * 16;
  const int col0 = wave * 16;
  const int m    = lane & 15;                // A row / B,C,D column within tile
  const int koff = (lane >> 4) << 1;         // 0 for lanes 0-15, 2 for lanes 16-31

  v8f c = {0.f, 0.f, 0.f, 0.f, 0.f, 0.f, 0.f, 0.f};

  const float* hrow = h   + (size_t)(row0 + m) * HDIM;
  const float* arow = agg + (size_t)(row0 + m) * HDIM;

  for (int k = 0; k < HDIM; k += 4) {
    v2f av, bv;
    // A from h (16x4), B from Wroot (4x16)
    av.x = hrow[k + koff];  av.y = hrow[k + koff + 1];
    bv.x = Wr[(size_t)(k + koff) * HDIM + col0 + m];
    bv.y = Wr[(size_t)(k + koff + 1) * HDIM + col0 + m];
    c = __builtin_amdgcn_wmma_f32_16x16x4_f32(false, av, false, bv, (short)0, c, false, false);
    // A from agg, B from Wrel
    av.x = arow[k + koff];  av.y = arow[k + koff + 1];
    bv.x = Wn[(size_t)(k + koff) * HDIM + col0 + m];
    bv.y = Wn[(size_t)(k + koff + 1) * HDIM + col0 + m];
    c = __builtin_amdgcn_wmma_f32_16x16x4_f32(false, av, false, bv, (short)0, c, false, false);
  }

  // D layout: element r <-> row (r + 8*(lane>>4)), col (lane&15)
  const int hi   = (lane >> 4) * 8;
  const float bc = bias[col0 + m];
#pragma unroll
  for (int r = 0; r < 8; ++r) {
    int row = row0 + hi + r;
    float v = c[r] + bc;
    v = v > 0.f ? v : 0.f;
    v *= maskf[row];
    out[(size_t)row * HDIM + col0 + m] = v;
  }
}

// ---------------- ||p|| + eps ------------------------------------------------
__global__ void pnorm_kernel(const float* __restrict__ p, float* __restrict__ pnorm) {
  __shared__ float red[HDIM];
  int t = threadIdx.x;
  float v = p[t];
  red[t] = v * v;
  __syncthreads();
  for (int s = HDIM / 2; s > 0; s >>= 1) { if (t < s) red[t] += red[t + s]; __syncthreads(); }
  if (t == 0) pnorm[0] = sqrtf(red[0]) + 1e-8f;
}

// ---------------- score[i] = h[i].p / (||p||+eps) ----------------------------
__global__ void score_kernel(const float* __restrict__ h, const float* __restrict__ p,
                             const float* __restrict__ pnorm, float* __restrict__ score) {
  __shared__ float red[HDIM];
  int i = blockIdx.x, t = threadIdx.x;
  red[t] = h[(size_t)i * HDIM + t] * p[t];
  __syncthreads();
  for (int s = HDIM / 2; s > 0; s >>= 1) { if (t < s) red[t] += red[t + s]; __syncthreads(); }
  if (t == 0) score[i] = red[0] / pnorm[0];
}

// ---------------- top-k pool per graph (exact lexsort tie-breaking) ----------
__global__ void pool_kernel(const int* __restrict__ starts, const int* __restrict__ totals,
                            const float* __restrict__ score,
                            float* __restrict__ maskf, float* __restrict__ h) {
  int g = blockIdx.x;
  int s = starts[g], t = totals[g];
  __shared__ float ssc[MAXSEG];
  __shared__ unsigned char salive[MAXSEG];
  __shared__ float sfac[MAXSEG];
  __shared__ int s_alive;

  for (int i = threadIdx.x; i < t && i < MAXSEG; i += blockDim.x) {
    ssc[i] = score[s + i];
    salive[i] = (maskf[s + i] > 0.5f) ? 1 : 0;
  }
  if (threadIdx.x == 0) s_alive = 0;
  __syncthreads();

  int la = 0;
  for (int i = threadIdx.x; i < t; i += blockDim.x)
    la += ((i < MAXSEG) ? (salive[i] != 0) : (maskf[s + i] > 0.5f)) ? 1 : 0;
  atomicAdd(&s_alive, la);
  __syncthreads();

  int alive = s_alive;
  int k = (alive + 1) >> 1;                  // ceil(0.5*alive)

  for (int i = threadIdx.x; i < t; i += blockDim.x) {
    float sci = (i < MAXSEG) ? ssc[i] : score[s + i];
    bool  ai  = (i < MAXSEG) ? (salive[i] != 0) : (maskf[s + i] > 0.5f);
    int rank = 0;
    if (ai) {
      for (int j = 0; j < t; ++j) {
        float scj = (j < MAXSEG) ? ssc[j] : score[s + j];
        bool  aj  = (j < MAXSEG) ? (salive[j] != 0) : (maskf[s + j] > 0.5f);
        if (aj && (scj > sci || (scj == sci && j < i))) rank++;
      }
    }
    bool keep = ai && (rank < k);
    float fac = keep ? tanhf(sci) : 0.0f;
    maskf[s + i] = keep ? 1.0f : 0.0f;
    if (i < MAXSEG) {
      sfac[i] = fac;
    } else {
      for (int f = 0; f < HDIM; ++f) h[(size_t)(s + i) * HDIM + f] *= fac;
    }
  }
  __syncthreads();

  int tl = t < MAXSEG ? t : MAXSEG;
  for (int idx = threadIdx.x; idx < tl * HDIM; idx += blockDim.x) {
    int i = idx >> 7, f = idx & (HDIM - 1);
    h[(size_t)(s + i) * HDIM + f] *= sfac[i];
  }
}

// ---------------- readout: [gmp || gap] per graph into z columns -------------
__global__ void readout_kernel(const int* __restrict__ starts, const int* __restrict__ totals,
                               const float* __restrict__ h, const float* __restrict__ maskf,
                               float* __restrict__ z, int colOff) {
  int g = blockIdx.x, f = threadIdx.x;       // 128 threads
  int s = starts[g], t = totals[g];
  float mx = -INFINITY, sum = 0.f;
  int cnt = 0;
  for (int i = 0; i < t; ++i) {
    if (maskf[s + i] > 0.5f) {
      float v = h[(size_t)(s + i) * HDIM + f];
      sum += v;
      mx = fmaxf(mx, v);
      cnt++;
    }
  }
  float mean = sum / (float)(cnt > 0 ? cnt : 1);
  float mxo  = (cnt > 0) ? mx : 0.f;
  z[(size_t)g * 768 + colOff + f]        = mxo;
  z[(size_t)g * 768 + colOff + HDIM + f] = mean;
}

// ---------------- classifier MLP ---------------------------------------------
__global__ void mlp1_kernel(const float* __restrict__ z, const float* __restrict__ W,
                            const float* __restrict__ b, float* __restrict__ o) {
  int row = blockIdx.x, j = threadIdx.x;     // 128 threads
  float acc = b[j];
  const float* zr = z + (size_t)row * 768;
  for (int k = 0; k < 768; ++k) acc = fmaf(zr[k], W[(size_t)k * HDIM + j], acc);
  o[(size_t)row * HDIM + j] = fmaxf(acc, 0.f);
}

__global__ void mlp2_kernel(const float* __restrict__ a, const float* __restrict__ W,
                            const float* __restrict__ b, float* __restrict__ o) {
  int row = blockIdx.x, j = threadIdx.x;     // 64 threads
  float acc = b[j];
  const float* ar = a + (size_t)row * HDIM;
  for (int k = 0; k < HDIM; ++k) acc = fmaf(ar[k], W[(size_t)k * 64 + j], acc);
  o[(size_t)row * 64 + j] = fmaxf(acc, 0.f);
}

__global__ void mlp3_kernel(const float* __restrict__ a, const float* __restrict__ W,
                            const float* __restrict__ b, float* __restrict__ out) {
  int row = blockIdx.x;
  __shared__ float lg[NCLASS];
  int c = threadIdx.x;                       // 32 threads, 10 active
  if (c < NCLASS) {
    float acc = b[c];
    const float* ar = a + (size_t)row * 64;
    for (int k = 0; k < 64; ++k) acc = fmaf(ar[k], W[(size_t)k * NCLASS + c], acc);
    lg[c] = acc;
  }
  __syncthreads();
  if (c < NCLASS) {
    float mx = lg[0];
    for (int q = 1; q < NCLASS; ++q) mx = fmaxf(mx, lg[q]);
    float se = 0.f;
    for (int q = 0; q < NCLASS; ++q) se += expf(lg[q] - mx);
    out[(size_t)row * NCLASS + c] = lg[c] - mx - logf(se);
  }
}

// -----------------------------------------------------------------------------
extern "C" void kernel_launch(void* const* d_in, const int* in_sizes, int n_in,
                              void* d_out, int out_size, void* d_ws, size_t ws_size,
                              hipStream_t stream) {
  const float* x     = (const float*)d_in[0];
  const int*   ei    = (const int*)d_in[1];
  const int*   batch = (const int*)d_in[2];
  const float* Wroot[3] = {(const float*)d_in[3], (const float*)d_in[6], (const float*)d_in[9]};
  const float* Wrel[3]  = {(const float*)d_in[4], (const float*)d_in[7], (const float*)d_in[10]};
  const float* bb[3]    = {(const float*)d_in[5], (const float*)d_in[8], (const float*)d_in[11]};
  const float* pv[3]    = {(const float*)d_in[12], (const float*)d_in[13], (const float*)d_in[14]};
  const float* Wl1 = (const float*)d_in[15]; const float* bl1 = (const float*)d_in[16];
  const float* Wl2 = (const float*)d_in[17]; const float* bl2 = (const float*)d_in[18];
  const float* Wl3 = (const float*)d_in[19]; const float* bl3 = (const float*)d_in[20];

  char* ws = (char*)d_ws;
  size_t off = 0;
  auto alloc = [&](size_t bytes) -> void* {
    void* p = ws + off;
    off += (bytes + 255) & ~(size_t)255;
    return p;
  };
  float* agg    = (float*)alloc((size_t)N_NODES * HDIM * 4);
  float* hA     = (float*)alloc((size_t)N_NODES * HDIM * 4);
  float* hB     = (float*)alloc((size_t)N_NODES * HDIM * 4);
  float* score  = (float*)alloc((size_t)N_NODES * 4);
  float* maskf  = (float*)alloc((size_t)N_NODES * 4);
  float* pnorm  = (float*)alloc(256);
  int*   starts = (int*)alloc(NGRAPH * 4);
  int*   totals = (int*)alloc(NGRAPH * 4);
  float* z      = (float*)alloc((size_t)NGRAPH * 768 * 4);
  float* a1     = (float*)alloc((size_t)NGRAPH * HDIM * 4);
  float* a2     = (float*)alloc((size_t)NGRAPH * 64 * 4);
  int*   deg    = (int*)alloc((size_t)N_NODES * 4);        // also reused as fill cursor
  int*   rowptr = (int*)alloc((size_t)(N_NODES + 1) * 4);
  int*   csr_src= (int*)alloc((size_t)N_EDGES * 4);

  // ---- one-time per launch: segment bounds, mask init, CSR-by-dst build ----
  seg_bounds_kernel<<<1, NGRAPH, 0, stream>>>(batch, starts, totals);
  init_mask_kernel<<<(N_NODES + 255) / 256, 256, 0, stream>>>(maskf);

  hipMemsetAsync(deg, 0, (size_t)N_NODES * 4, stream);
  degree_kernel<<<(N_EDGES + 255) / 256, 256, 0, stream>>>(ei, deg);
  scan_kernel<<<1, 1024, 0, stream>>>(deg, rowptr);
  copy_pos_kernel<<<(N_NODES + 255) / 256, 256, 0, stream>>>(rowptr, deg);   // deg -> cursor
  fill_kernel<<<(N_EDGES + 255) / 256, 256, 0, stream>>>(ei, deg, csr_src);
  sortbucket_kernel<<<(N_NODES + 255) / 256, 256, 0, stream>>>(rowptr, csr_src);

  // ---- 3x (conv -> pool -> readout) ----
  const float* hin = x;
  float* bufs[2] = {hA, hB};
  for (int l = 0; l < 3; ++l) {
    float* hout = bufs[l & 1];
    gather_kernel<<<N_NODES, HDIM, 0, stream>>>(rowptr, csr_src, hin, agg);
    conv_wmma_kernel<<<N_NODES / 16, 256, 0, stream>>>(hin, agg, Wroot[l], Wrel[l], bb[l], maskf, hout);
    pnorm_kernel<<<1, HDIM, 0, stream>>>(pv[l], pnorm);
    score_kernel<<<N_NODES, HDIM, 0, stream>>>(hout, pv[l], pnorm, score);
    pool_kernel<<<NGRAPH, 256, 0, stream>>>(starts, totals, score, maskf, hout);
    readout_kernel<<<NGRAPH, HDIM, 0, stream>>>(starts, totals, hout, maskf, z, l * 256);
    hin = hout;
  }

  mlp1_kernel<<<NGRAPH, HDIM, 0, stream>>>(z, Wl1, bl1, a1);
  mlp2_kernel<<<NGRAPH, 64, 0, stream>>>(a1, Wl2, bl2, a2);
  mlp3_kernel<<<NGRAPH, 32, 0, stream>>>(a2, Wl3, bl3, (float*)d_out);
}